// ProbUCELossEF_CE_15444702397047
// MI455X (gfx1250) — compile-verified
//
#include <hip/hip_runtime.h>
#include <hip/hip_bf16.h>
#include <math.h>

typedef float v2f __attribute__((ext_vector_type(2)));
typedef float v8f __attribute__((ext_vector_type(8)));

#define NTGT  32
#define NBINS 15

static __device__ __forceinline__ int bperm_i(int srcLane, int v) {
  return __builtin_amdgcn_ds_bpermute(srcLane << 2, v);
}
static __device__ __forceinline__ float bperm_f(int srcLane, float v) {
  return __int_as_float(__builtin_amdgcn_ds_bpermute(srcLane << 2, __float_as_int(v)));
}
// select element r (0..7) of an 8-wide f32 accumulator (constant-index extracts + cndmask chain)
static __device__ __forceinline__ float sel8(v8f d, int r) {
  float v = d[0];
  v = (r == 1) ? d[1] : v;
  v = (r == 2) ? d[2] : v;
  v = (r == 3) ? d[3] : v;
  v = (r == 4) ? d[4] : v;
  v = (r == 5) ? d[5] : v;
  v = (r == 6) ? d[6] : v;
  v = (r == 7) ? d[7] : v;
  return v;
}

// ---------------- zero scratch ----------------
__global__ void k_zero(unsigned int* p, int nwords) {
  int i = blockIdx.x * blockDim.x + threadIdx.x;
  if (i < nwords) p[i] = 0u;
}

// ---------------- pass 1: per-row H2 + err via WMMA row-sums ----------------
// Each wave handles a 16x16 tile (16 rows). A-matrix layout for V_WMMA_F32_16X16X4_F32:
// lanes 0-15 hold rows, K = 4c..4c+1 ; lanes 16-31 hold K = 4c+2..4c+3.
__global__ __launch_bounds__(256) void k_stats(const float* __restrict__ logits,
                                               const int* __restrict__ labels,
                                               unsigned int* __restrict__ packed,
                                               unsigned int* __restrict__ hist1,
                                               int n) {
  const int lane = threadIdx.x & 31;
  const int half = lane >> 4;           // 0: K even pair, 1: K odd pair
  const int r    = lane & 15;           // tile row this lane loads
  const long long wgid   = (long long)blockIdx.x * (blockDim.x >> 5) + (threadIdx.x >> 5);
  const long long nwaves = (long long)gridDim.x * (blockDim.x >> 5);
  const long long ntiles = (long long)(n >> 4);

  for (long long tile = wgid; tile < ntiles; tile += nwaves) {
    const float* rowp = logits + ((tile << 4) + r) * 16 + half * 2;
    v2f x0 = *(const v2f*)(rowp + 0);
    v2f x1 = *(const v2f*)(rowp + 4);
    v2f x2 = *(const v2f*)(rowp + 8);
    v2f x3 = *(const v2f*)(rowp + 12);

    // local max / first-occurrence argmax over this lane's 8 columns
    const int cb = half * 2;
    float m = x0.x; int am = cb;
    if (x0.y > m) { m = x0.y; am = cb + 1;  }
    if (x1.x > m) { m = x1.x; am = cb + 4;  }
    if (x1.y > m) { m = x1.y; am = cb + 5;  }
    if (x2.x > m) { m = x2.x; am = cb + 8;  }
    if (x2.y > m) { m = x2.y; am = cb + 9;  }
    if (x3.x > m) { m = x3.x; am = cb + 12; }
    if (x3.y > m) { m = x3.y; am = cb + 13; }
    // combine with partner lane (same row, other K-half); tie -> smaller index
    float pm  = bperm_f(lane ^ 16, m);
    int   pam = bperm_i(lane ^ 16, am);
    if (pm > m || (pm == m && pam < am)) { m = pm; am = pam; }

    // exp(x - m) and squares
    v2f e0, e1, e2, e3, q0, q1, q2, q3;
    e0.x = __expf(x0.x - m); e0.y = __expf(x0.y - m);
    e1.x = __expf(x1.x - m); e1.y = __expf(x1.y - m);
    e2.x = __expf(x2.x - m); e2.y = __expf(x2.y - m);
    e3.x = __expf(x3.x - m); e3.y = __expf(x3.y - m);
    q0.x = e0.x * e0.x; q0.y = e0.y * e0.y;
    q1.x = e1.x * e1.x; q1.y = e1.y * e1.y;
    q2.x = e2.x * e2.x; q2.y = e2.y * e2.y;
    q3.x = e3.x * e3.x; q3.y = e3.y * e3.y;

    // row sums via WMMA against all-ones B (B layout irrelevant for ones)
    v2f bones; bones.x = 1.0f; bones.y = 1.0f;
    v8f s1 = {0.f,0.f,0.f,0.f,0.f,0.f,0.f,0.f};
    v8f s2 = {0.f,0.f,0.f,0.f,0.f,0.f,0.f,0.f};
    s1 = __builtin_amdgcn_wmma_f32_16x16x4_f32(false, e0, false, bones, (short)0, s1, false, false);
    s1 = __builtin_amdgcn_wmma_f32_16x16x4_f32(false, e1, false, bones, (short)0, s1, false, false);
    s1 = __builtin_amdgcn_wmma_f32_16x16x4_f32(false, e2, false, bones, (short)0, s1, false, false);
    s1 = __builtin_amdgcn_wmma_f32_16x16x4_f32(false, e3, false, bones, (short)0, s1, false, false);
    s2 = __builtin_amdgcn_wmma_f32_16x16x4_f32(false, q0, false, bones, (short)0, s2, false, false);
    s2 = __builtin_amdgcn_wmma_f32_16x16x4_f32(false, q1, false, bones, (short)0, s2, false, false);
    s2 = __builtin_amdgcn_wmma_f32_16x16x4_f32(false, q2, false, bones, (short)0, s2, false, false);
    s2 = __builtin_amdgcn_wmma_f32_16x16x4_f32(false, q3, false, bones, (short)0, s2, false, false);

    // D layout: lanes 0-15 hold rows 0-7 in d[0..7]; lanes 16-31 hold rows 8-15.
    // Writers: lanes 0-7 -> rows 0-7, lanes 16-23 -> rows 8-15.
    const int rowl = (lane & 7) | ((lane & 16) >> 1);
    const int amr  = bperm_i(rowl, am);     // argmax of row rowl (held by lane rowl)
    if ((lane & 8) == 0) {
      float v1 = sel8(s1, lane & 7);
      float v2 = sel8(s2, lane & 7);
      long long grow = (tile << 4) + rowl;
      unsigned int err = (amr != labels[grow]) ? 1u : 0u;
      float sump2 = __fdividef(v2, v1 * v1);   // fast rcp path; feeds log2(x + 1e-12)
      float H2 = -__log2f(sump2 + 1e-12f);
      H2 = fmaxf(H2, 0.0f);                    // keep sign bit free for err
      unsigned int bits = __float_as_uint(H2) | (err << 31);
      packed[grow] = bits;
      atomicAdd(&hist1[(bits & 0x7fffffffu) >> 16], 1u);
    }
  }
}

// ---------------- radix-select level 1 scan ----------------
__global__ void k_scan1(const unsigned int* __restrict__ hist1,
                        unsigned int* tb1, unsigned int* tr1, int n) {
  if (threadIdx.x != 0 || blockIdx.x != 0) return;
  long long ranks[NTGT];
  for (int i = 0; i <= 15; ++i) {
    double pos = (double)i * (double)(n - 1) / 15.0;
    long long k0 = (long long)pos;
    long long k1 = k0 + 1; if (k1 > (long long)n - 1) k1 = (long long)n - 1;
    ranks[2 * i]     = k0;
    ranks[2 * i + 1] = k1;
  }
  long long cum = 0; int ti = 0;
  for (int b = 0; b < 32768 && ti < NTGT; ++b) {
    long long nc = cum + (long long)hist1[b];
    while (ti < NTGT && ranks[ti] < nc) {
      tb1[ti] = (unsigned int)b;
      tr1[ti] = (unsigned int)(ranks[ti] - cum);
      ++ti;
    }
    cum = nc;
  }
}

// ---------------- radix-select level 2 histogram ----------------
__global__ __launch_bounds__(256) void k_hist2(const unsigned int* __restrict__ packed,
                                               const unsigned int* __restrict__ tb1,
                                               unsigned int* __restrict__ hist2, int n) {
  __shared__ unsigned int sb[NTGT];
  if (threadIdx.x < NTGT) sb[threadIdx.x] = tb1[threadIdx.x];
  __syncthreads();
  long long stride = (long long)gridDim.x * blockDim.x;
  for (long long i = (long long)blockIdx.x * blockDim.x + threadIdx.x; i < n; i += stride) {
    unsigned int key = packed[i] & 0x7fffffffu;
    unsigned int hi  = key >> 16;
    unsigned int mid = (key >> 8) & 0xffu;
    #pragma unroll
    for (int t = 0; t < NTGT; ++t)
      if (sb[t] == hi) atomicAdd(&hist2[t * 256 + mid], 1u);
  }
}

__global__ void k_scan2(const unsigned int* __restrict__ hist2,
                        const unsigned int* __restrict__ tr1,
                        unsigned int* tb2, unsigned int* tr2) {
  int t = threadIdx.x;
  if (t >= NTGT || blockIdx.x != 0) return;
  unsigned int r = tr1[t], cum = 0;
  for (int j = 0; j < 256; ++j) {
    unsigned int c = hist2[t * 256 + j];
    if (r < cum + c) { tb2[t] = (unsigned int)j; tr2[t] = r - cum; return; }
    cum += c;
  }
  tb2[t] = 255u; tr2[t] = 0u;
}

// ---------------- radix-select level 3 histogram ----------------
__global__ __launch_bounds__(256) void k_hist3(const unsigned int* __restrict__ packed,
                                               const unsigned int* __restrict__ tb1,
                                               const unsigned int* __restrict__ tb2,
                                               unsigned int* __restrict__ hist3, int n) {
  __shared__ unsigned int sp[NTGT];
  if (threadIdx.x < NTGT) sp[threadIdx.x] = (tb1[threadIdx.x] << 8) | tb2[threadIdx.x];
  __syncthreads();
  long long stride = (long long)gridDim.x * blockDim.x;
  for (long long i = (long long)blockIdx.x * blockDim.x + threadIdx.x; i < n; i += stride) {
    unsigned int key  = packed[i] & 0x7fffffffu;
    unsigned int pref = key >> 8;
    unsigned int lo   = key & 0xffu;
    #pragma unroll
    for (int t = 0; t < NTGT; ++t)
      if (sp[t] == pref) atomicAdd(&hist3[t * 256 + lo], 1u);
  }
}

__global__ void k_scan3(const unsigned int* __restrict__ hist3,
                        const unsigned int* __restrict__ tr2,
                        const unsigned int* __restrict__ tb1,
                        const unsigned int* __restrict__ tb2,
                        float* edges, int n) {
  __shared__ float vals[NTGT];
  int t = threadIdx.x;
  if (t < NTGT) {
    unsigned int r = tr2[t], cum = 0, b3 = 255u;
    for (int j = 0; j < 256; ++j) {
      unsigned int c = hist3[t * 256 + j];
      if (r < cum + c) { b3 = (unsigned int)j; break; }
      cum += c;
    }
    vals[t] = __uint_as_float((tb1[t] << 16) | (tb2[t] << 8) | b3);
  }
  __syncthreads();
  if (t == 0 && blockIdx.x == 0) {
    for (int i = 0; i <= 15; ++i) {
      double pos  = (double)i * (double)(n - 1) / 15.0;
      double frac = pos - (double)(long long)pos;
      double v0 = (double)vals[2 * i], v1 = (double)vals[2 * i + 1];
      float e = (float)(v0 + frac * (v1 - v0));
      if (i == 15) e += 1e-6f;
      edges[i] = e;
    }
  }
}

// ---------------- binned accumulation (deterministic fixed-point) ----------------
__global__ __launch_bounds__(256) void k_bin(const unsigned int* __restrict__ packed,
                                             const float* __restrict__ edges,
                                             unsigned long long* __restrict__ binH2,
                                             unsigned int* __restrict__ binCnt,
                                             unsigned int* __restrict__ binErr, int n) {
  __shared__ float se[16];
  __shared__ unsigned long long sh[NBINS];
  __shared__ unsigned int sc[NBINS], sv[NBINS];
  if (threadIdx.x < 16) se[threadIdx.x] = edges[threadIdx.x];
  if (threadIdx.x < NBINS) { sh[threadIdx.x] = 0ull; sc[threadIdx.x] = 0u; sv[threadIdx.x] = 0u; }
  __syncthreads();
  long long stride = (long long)gridDim.x * blockDim.x;
  for (long long i = (long long)blockIdx.x * blockDim.x + threadIdx.x; i < n; i += stride) {
    unsigned int p = packed[i];
    float H2 = __uint_as_float(p & 0x7fffffffu);
    unsigned int err = p >> 31;
    int b = -1;
    for (int j = 0; j < NBINS; ++j)
      if (H2 > se[j] && H2 <= se[j + 1]) { b = j; break; }
    if (b >= 0) {
      unsigned long long fx = (unsigned long long)(H2 * 16777216.0f + 0.5f); // 2^24 fixed point
      atomicAdd(&sh[b], fx);
      atomicAdd(&sc[b], 1u);
      atomicAdd(&sv[b], err);
    }
  }
  __syncthreads();
  if (threadIdx.x < NBINS) {
    atomicAdd(&binH2[threadIdx.x], sh[threadIdx.x]);
    atomicAdd(&binCnt[threadIdx.x], sc[threadIdx.x]);
    atomicAdd(&binErr[threadIdx.x], sv[threadIdx.x]);
  }
}

// ---------------- final reduction ----------------
__global__ void k_final(const unsigned long long* __restrict__ binH2,
                        const unsigned int* __restrict__ binCnt,
                        const unsigned int* __restrict__ binErr, float* out) {
  if (threadIdx.x != 0 || blockIdx.x != 0) return;
  float sum = 0.0f;
  for (int j = 0; j < NBINS; ++j) {
    float gap = 0.0f;
    unsigned int c = binCnt[j];
    if (c > 0u) {
      float u  = (float)((double)binH2[j] / 16777216.0 / (double)c);
      float eb = (float)binErr[j] / (float)c;
      float inner = 2.0f * exp2f(-u) - 1.0f;
      float risk  = (inner > 0.0f) ? 0.5f * (1.0f - sqrtf(inner)) : 0.5f;
      gap = fabsf(eb - risk);
    }
    sum += gap;
  }
  out[0] = sum / 15.0f;
}

extern "C" void kernel_launch(void* const* d_in, const int* in_sizes, int n_in,
                              void* d_out, int out_size, void* d_ws, size_t ws_size,
                              hipStream_t stream) {
  const float* logits = (const float*)d_in[0];
  const int*   labels = (const int*)d_in[1];
  float*       out    = (float*)d_out;
  const int n = in_sizes[1];                 // number of rows (labels count)

  char* ws = (char*)d_ws;
  size_t b0 = (((size_t)n * 4) + 255) & ~(size_t)255;
  unsigned int* packed = (unsigned int*)ws;
  unsigned int* hist1  = (unsigned int*)(ws + b0);                    // 32768 u32
  unsigned int* hist2  = (unsigned int*)(ws + b0 + 131072);           // 32*256 u32
  unsigned int* hist3  = (unsigned int*)(ws + b0 + 131072 + 32768);   // 32*256 u32
  char* tsel = ws + b0 + 196608;                                      // 1 KB
  unsigned int* tb1 = (unsigned int*)(tsel + 0);
  unsigned int* tr1 = (unsigned int*)(tsel + 128);
  unsigned int* tb2 = (unsigned int*)(tsel + 256);
  unsigned int* tr2 = (unsigned int*)(tsel + 384);
  float*        edges = (float*)(tsel + 512);
  char* bins = ws + b0 + 197632;                                      // 256 B
  unsigned long long* binH2 = (unsigned long long*)(bins);
  unsigned int* binCnt = (unsigned int*)(bins + 128);
  unsigned int* binErr = (unsigned int*)(bins + 192);

  const int zerowords = 197888 / 4;   // hist1..bins inclusive
  k_zero<<<(zerowords + 255) / 256, 256, 0, stream>>>(hist1, zerowords);

  k_stats<<<2048, 256, 0, stream>>>(logits, labels, packed, hist1, n);
  k_scan1<<<1, 32, 0, stream>>>(hist1, tb1, tr1, n);
  k_hist2<<<2048, 256, 0, stream>>>(packed, tb1, hist2, n);
  k_scan2<<<1, 32, 0, stream>>>(hist2, tr1, tb2, tr2);
  k_hist3<<<2048, 256, 0, stream>>>(packed, tb1, tb2, hist3, n);
  k_scan3<<<1, 32, 0, stream>>>(hist3, tr2, tb1, tb2, edges, n);
  k_bin<<<2048, 256, 0, stream>>>(packed, edges, binH2, binCnt, binErr, n);
  k_final<<<1, 1, 0, stream>>>(binH2, binCnt, binErr, out);
}